// Dynamic_Filter_Layer_89172110999960
// MI455X (gfx1250) — compile-verified
//
#include <hip/hip_runtime.h>

typedef __attribute__((ext_vector_type(2))) float v2f;
typedef __attribute__((ext_vector_type(4))) float v4f;
typedef __attribute__((ext_vector_type(8))) float v8f;

#define B_  8
#define H_  256
#define W_  256
#define C_  64
#define HO_ 252
#define WO_ 252

// Dynamic filter layer as banded fp32 WMMA, two output rows per wave.
// Out(16j x 16c) for rows i and i+1 = sum over di of (16x20 banded flow strip)
// x (20x16 x-slab), as 2 x 25 chained V_WMMA_F32_16X16X4_F32 sharing B operands:
// x row i+r feeds row-i's di=r chain and row-(i+1)'s di=r-1 chain -> 60 B loads
// instead of 100, and two independent WMMA chains interleave in the matrix pipe.
//  - A (banded flow) scattered once into LDS in WMMA fragment order (band zeros
//    baked in); index math shared between the two flow rows.
//  - B (x) read directly from global with compile-time immediate offsets.
//  - Last j tile overlapped (j0 = Wo-16) so every access is in-bounds: no masks,
//    no clamps, no guarded stores; overlap outputs recomputed identically.
__global__ __launch_bounds__(128) void dfl_wmma_kernel(
    const float* __restrict__ x, const float* __restrict__ flow,
    float* __restrict__ out)
{
    __shared__ __align__(16) float Alds[2 * 1600];   // A fragments for rows i0, i0+1

    const int tj = blockIdx.x;        // j tile 0..15
    const int i0 = blockIdx.y * 2;    // 0,2,..,250
    const int b  = blockIdx.z;        // 0..7
    int j0 = tj * 16;
    if (j0 > WO_ - 16) j0 = WO_ - 16; // overlap final tile -> no edge cases
    const int t = threadIdx.x;

    // ---- zero A fragments (b128 stores) ------------------------------------
    {
        const v4f z = {};
#pragma unroll
        for (int r = 0; r < 7; ++r) {
            const int idx4 = t + r * 128;          // 800 float4 cells
            if (idx4 < 800) *(v4f*)(Alds + idx4 * 4) = z;
        }
    }
    __syncthreads();

    // ---- scatter both flow rows into banded-A WMMA fragment layout ----------
    // A[jj, p] = flow[jj, di*5+dj] at p = jj+dj; fragment float index:
    //   ((di*5 + (p>>2))*2 + ((p>>1)&1))*32 + jj*2 + (p&1)
    // fi depends only on idx -> computed once, reused for both rows.
    {
        const float* frow0 = flow + ((size_t)(b * HO_ + i0) * WO_ + j0) * 25;
        const float* frow1 = frow0 + (size_t)WO_ * 25;
#pragma unroll
        for (int rr = 0; rr < 4; ++rr) {
            const int idx = t + rr * 128;          // 0..399, coalesced reads
            if (idx < 400) {
                const int jj  = idx / 25;
                const int tap = idx - jj * 25;
                const int di  = tap / 5;
                const int dj  = tap - di * 5;
                const int p   = jj + dj;
                const int fi  = ((di * 5 + (p >> 2)) * 2 + ((p >> 1) & 1)) * 32
                                + jj * 2 + (p & 1);
                Alds[fi]        = frow0[idx];
                Alds[1600 + fi] = frow1[idx];
            }
        }
    }
    __syncthreads();

    // ---- 50 WMMAs in two interleaved chains, B shared -----------------------
    const int wave = t >> 5;                 // channel tile 0..3
    const int lane = t & 31;
    const int half = lane >> 4;              // 0/1
    const int M    = lane & 15;              // A row (j) == B column (channel)
    const int c0   = wave * 16;

    const float* Ab0 = Alds + half * 32 + M * 2;   // + (di*5+ch)*64 floats (imm)
    const float* Ab1 = Ab0 + 1600;
    // B[p, M] = x[b, i0+r, j0+p, c0+M], p = ch*4 + 2*half + v; fold 2*half into base.
    const float* xb = x + (((size_t)(b * H_ + i0) * W_) + j0 + 2 * half) * C_ + c0 + M;

    v8f acc0 = {};
    v8f acc1 = {};
#pragma unroll
    for (int r = 0; r < 6; ++r) {            // x row i0 + r
#pragma unroll
        for (int ch = 0; ch < 5; ++ch) {
            const float b0 = xb[(size_t)(r * W_ + ch * 4) * C_];       // imm-offset
            const float b1 = xb[(size_t)(r * W_ + ch * 4 + 1) * C_];   // imm-offset
            const v2f bv = {b0, b1};
            if (r < 5) {                     // row i0 chain: di = r
                const v2f av = *(const v2f*)(Ab0 + (r * 5 + ch) * 64); // ds_load_b64
                acc0 = __builtin_amdgcn_wmma_f32_16x16x4_f32(
                    false, av, false, bv, (short)0, acc0, false, false);
            }
            if (r >= 1) {                    // row i0+1 chain: di = r-1
                const v2f av = *(const v2f*)(Ab1 + ((r - 1) * 5 + ch) * 64);
                acc1 = __builtin_amdgcn_wmma_f32_16x16x4_f32(
                    false, av, false, bv, (short)0, acc1, false, false);
            }
        }
    }

    // ---- store: vgpr v -> row jj = v + 8*half, channel c0 + M ---------------
    float* ob0 = out + (((size_t)(b * HO_ + i0) * WO_) + j0 + 8 * half) * C_ + c0 + M;
    float* ob1 = ob0 + (size_t)WO_ * C_;
#pragma unroll
    for (int v = 0; v < 8; ++v) {
        __builtin_nontemporal_store(acc0[v], ob0 + (size_t)v * C_);
        __builtin_nontemporal_store(acc1[v], ob1 + (size_t)v * C_);
    }
}

extern "C" void kernel_launch(void* const* d_in, const int* in_sizes, int n_in,
                              void* d_out, int out_size, void* d_ws, size_t ws_size,
                              hipStream_t stream) {
    const float* x    = (const float*)d_in[0];
    const float* flow = (const float*)d_in[1];
    // d_in[2] is ksize (int scalar) == 5, baked in.
    float* out = (float*)d_out;

    dim3 grid(16 /* j tiles, last overlapped */, HO_ / 2 /* i pairs */, B_);
    dim3 block(128, 1, 1);   // 4 wave32, one 16-channel tile each, 2 output rows
    dfl_wmma_kernel<<<grid, block, 0, stream>>>(x, flow, out);
}